// GraphConvolution_28999619182702
// MI455X (gfx1250) — compile-verified
//
#include <hip/hip_runtime.h>
#include <hip/hip_bf16.h>
#include <math.h>

typedef __attribute__((ext_vector_type(2))) float v2f;
typedef __attribute__((ext_vector_type(8))) float v8f;

#define NNODES 50000
#define NEDGES 800000
#define FDIM   128
#define S_STRIDE  132  // support tile LDS stride (floats): A-frag ds_load_b64 conflict-free
#define WP_STRIDE 288  // k-pair-interleaved weight LDS stride: B-frag ds_load_b64 conflict-free

// ---------------------------------------------------------------------------
// Generic zero kernel (float4 granularity)
// ---------------------------------------------------------------------------
__global__ void gcn_zero_kernel(float4* __restrict__ p, int n4) {
    int i = blockIdx.x * blockDim.x + threadIdx.x;
    if (i < n4) p[i] = make_float4(0.f, 0.f, 0.f, 0.f);
}

// ===========================================================================
// CSR-by-dst path (atomic-free accumulation): histogram -> scan -> fill ->
// per-row gather. Cuts 102.4M f32 atomics down to 1.6M int atomics.
// ===========================================================================
__global__ void gcn_count_kernel(const int* __restrict__ dst,
                                 int* __restrict__ counts) {
    int e = blockIdx.x * blockDim.x + threadIdx.x;
    if (e < NEDGES) atomicAdd(&counts[dst[e]], 1);
}

// Single-workgroup exclusive scan over 50000 counts (49 chunks of 1024,
// Hillis-Steele in LDS). Tiny: ~0.5K LDS ops total.
__global__ void __launch_bounds__(1024)
gcn_scan_kernel(const int* __restrict__ counts,
                int* __restrict__ offsets,
                int* __restrict__ cursors) {
    __shared__ int sdata[1024];
    const int tid = threadIdx.x;
    const int nChunks = (NNODES + 1023) / 1024;
    int running = 0;
    for (int ch = 0; ch < nChunks; ++ch) {
        int idx = ch * 1024 + tid;
        int v = (idx < NNODES) ? counts[idx] : 0;
        sdata[tid] = v;
        __syncthreads();
        #pragma unroll
        for (int off = 1; off < 1024; off <<= 1) {
            int t = (tid >= off) ? sdata[tid - off] : 0;
            __syncthreads();
            sdata[tid] += t;
            __syncthreads();
        }
        int excl = sdata[tid] - v;
        if (idx < NNODES) {
            offsets[idx] = running + excl;
            cursors[idx] = running + excl;
        }
        int total = sdata[1023];
        __syncthreads();            // protect sdata before next chunk
        running += total;
    }
    if (tid == 0) offsets[NNODES] = running;
}

__global__ void gcn_fill_kernel(const int* __restrict__ src,
                                const int* __restrict__ dst,
                                const float* __restrict__ ew,
                                int* __restrict__ cursors,
                                int2* __restrict__ pack) {
    int e = blockIdx.x * blockDim.x + threadIdx.x;
    if (e >= NEDGES) return;
    int slot = atomicAdd(&cursors[dst[e]], 1);
    pack[slot] = make_int2(src[e], __float_as_int(ew[e]));
}

// One wave32 per destination row; lane = 4 features. x rows (25.6 MB) live in
// the 192 MB L2, so the per-edge 512B gathers are L2 hits. Single float4
// store per row -- no atomics.
__global__ void gcn_gather_kernel(const float* __restrict__ x,
                                  const int* __restrict__ offsets,
                                  const int2* __restrict__ pack,
                                  float* __restrict__ hi) {
    int row  = (blockIdx.x * blockDim.x + threadIdx.x) >> 5;
    int lane = threadIdx.x & 31;
    if (row >= NNODES) return;
    int beg = offsets[row], end = offsets[row + 1];
    float4 acc = make_float4(0.f, 0.f, 0.f, 0.f);
    for (int i = beg; i < end; ++i) {
        int2 p = pack[i];                    // broadcast (all lanes same addr)
        float w = __int_as_float(p.y);
        float4 v = ((const float4*)(x + (size_t)p.x * FDIM))[lane];
        acc.x += w * v.x; acc.y += w * v.y;
        acc.z += w * v.z; acc.w += w * v.w;
    }
    ((float4*)(hi + (size_t)row * FDIM))[lane] = acc;
}

// ===========================================================================
// Fallback scatter path (if ws too small): f32 atomics, one wave per edge.
// ===========================================================================
__global__ void gcn_spmm_kernel(const float* __restrict__ x,
                                const float* __restrict__ ew,
                                const int*   __restrict__ src,
                                const int*   __restrict__ dst,
                                float*       __restrict__ hi) {
    int e    = (blockIdx.x * blockDim.x + threadIdx.x) >> 5;
    int lane = threadIdx.x & 31;
    if (e >= NEDGES) return;
    int   s = src[e];
    int   d = dst[e];
    float w = ew[e];
    float4 v = ((const float4*)(x + (size_t)s * FDIM))[lane];
    float* hr = hi + (size_t)d * FDIM + lane * 4;
    atomicAdd(hr + 0, w * v.x);
    atomicAdd(hr + 1, w * v.y);
    atomicAdd(hr + 2, w * v.z);
    atomicAdd(hr + 3, w * v.w);
}

// ===========================================================================
// Fused: support = (1-a)*hi + a*h0 ; out = th*(support@W) + (1-th)*support + x
// In place over hi == d_out (row-independent GEMM; each WG reads its own rows
// into LDS before writing them back). 4 waves/WG, 16 rows/wave, 8 16x16 f32
// accumulators, K-loop of V_WMMA_F32_16X16X4_F32.
// Weight stored k-pair interleaved so each B fragment is ONE aligned
// ds_load_b64 into an even VGPR pair (no v_dual_mov repacking).
// ===========================================================================
__global__ void __launch_bounds__(128)
gcn_fused_kernel(const float* __restrict__ x,
                 const float* __restrict__ h0,
                 const float* __restrict__ W,
                 const float* __restrict__ lamdap,
                 const float* __restrict__ alphap,
                 const int*   __restrict__ lp,
                 float*       __restrict__ out /* == hi */) {
    __shared__ float sW[64 * WP_STRIDE];    // 73728 B, k-pair interleaved
    __shared__ float sS[64 * S_STRIDE];     // 33792 B (WGP has 320 KB LDS)

    const int   tid     = threadIdx.x;
    const float alpha   = *alphap;
    const float theta   = logf((*lamdap) / (float)(*lp) + 1.0f);
    const int   rowBase = blockIdx.x * 64;

    // --- stage weight, interleaving k-pairs: sW[k/2][n*2 + (k&1)] = W[k][n]
    for (int i = tid; i < FDIM * (FDIM / 4); i += 128) {
        int r  = i >> 5;
        int c  = (i & 31) * 4;
        float4 wv = ((const float4*)(W + (size_t)r * FDIM))[i & 31];
        float* d = &sW[(r >> 1) * WP_STRIDE + c * 2 + (r & 1)];
        d[0] = wv.x; d[2] = wv.y; d[4] = wv.z; d[6] = wv.w;
    }
    // --- stage support tile [64x128], fusing the (1-a)*hi + a*h0 blend ---
    for (int i = tid; i < 64 * (FDIM / 4); i += 128) {
        int r   = i >> 5;
        int c4  = i & 31;
        int gr  = rowBase + r;
        int sr  = gr < NNODES ? gr : (NNODES - 1);
        float4 hv  = ((const float4*)(out + (size_t)sr * FDIM))[c4];  // hi
        float4 h0v = ((const float4*)(h0  + (size_t)sr * FDIM))[c4];
        float4 sv;
        sv.x = (1.f - alpha) * hv.x + alpha * h0v.x;
        sv.y = (1.f - alpha) * hv.y + alpha * h0v.y;
        sv.z = (1.f - alpha) * hv.z + alpha * h0v.z;
        sv.w = (1.f - alpha) * hv.w + alpha * h0v.w;
        *(float4*)&sS[r * S_STRIDE + c4 * 4] = sv;
    }
    __syncthreads();

    const int lane    = tid & 31;
    const int lhalf   = lane & 15;
    const int uhalf   = lane >> 4;
    const int rowTile = (tid >> 5) * 16;

    v8f acc[8] = {};

    for (int K0 = 0; K0 < FDIM; K0 += 4) {
        // A frag: lane<16 -> A[lhalf][K0..K0+1]; lane>=16 -> A[lhalf][K0+2..K0+3]
        v2f a = *(const v2f*)&sS[(rowTile + lhalf) * S_STRIDE + K0 + 2 * uhalf];
        #pragma unroll
        for (int nt = 0; nt < 8; ++nt) {
            // B frag: {W[K0+2u][n], W[K0+2u+1][n]} = one b64 from pair row K0/2+u
            v2f b = *(const v2f*)&sW[(K0 / 2 + uhalf) * WP_STRIDE
                                     + (nt * 16 + lhalf) * 2];
            acc[nt] = __builtin_amdgcn_wmma_f32_16x16x4_f32(
                false, a, false, b, (short)0, acc[nt], false, false);
        }
    }

    const float omt = 1.0f - theta;
    #pragma unroll
    for (int nt = 0; nt < 8; ++nt) {
        #pragma unroll
        for (int vi = 0; vi < 8; ++vi) {
            int r  = rowTile + vi + 8 * uhalf;
            int c  = nt * 16 + lhalf;
            int gr = rowBase + r;
            if (gr < NNODES) {
                float sup = sS[r * S_STRIDE + c];
                out[(size_t)gr * FDIM + c] =
                    theta * acc[nt][vi] + omt * sup + x[(size_t)gr * FDIM + c];
            }
        }
    }
}

// ---------------------------------------------------------------------------
extern "C" void kernel_launch(void* const* d_in, const int* in_sizes, int n_in,
                              void* d_out, int out_size, void* d_ws, size_t ws_size,
                              hipStream_t stream) {
    const float* x      = (const float*)d_in[0];
    const float* h0     = (const float*)d_in[1];
    const float* edge_w = (const float*)d_in[2];
    const float* weight = (const float*)d_in[3];
    const float* lamda  = (const float*)d_in[4];
    const float* alpha  = (const float*)d_in[5];
    const int*   esrc   = (const int*)d_in[6];
    const int*   edst   = (const int*)d_in[7];
    const int*   lval   = (const int*)d_in[8];
    float* out = (float*)d_out;

    const int eBlocks = (NEDGES + 255) / 256;

    // Workspace layout (CSR path): [pack int2 x E][counts][offsets N+1][cursors]
    const size_t packBytes = (size_t)NEDGES * 8;
    const size_t needWs    = packBytes + (size_t)(NNODES + (NNODES + 1) + NNODES) * 4;

    if (ws_size >= needWs) {
        char* base    = (char*)d_ws;
        int2* pack    = (int2*)base;
        int*  counts  = (int*)(base + packBytes);
        int*  offsets = counts + NNODES;
        int*  cursors = offsets + (NNODES + 1);

        // zero counts (50000 ints == 12500 float4)
        gcn_zero_kernel<<<(NNODES / 4 + 255) / 256, 256, 0, stream>>>(
            (float4*)counts, NNODES / 4);
        gcn_count_kernel<<<eBlocks, 256, 0, stream>>>(edst, counts);
        gcn_scan_kernel<<<1, 1024, 0, stream>>>(counts, offsets, cursors);
        gcn_fill_kernel<<<eBlocks, 256, 0, stream>>>(esrc, edst, edge_w,
                                                     cursors, pack);
        // atomic-free SpMM: one wave per row, writes hi (== d_out) densely
        {
            long long thr = (long long)NNODES * 32;
            gcn_gather_kernel<<<(int)((thr + 255) / 256), 256, 0, stream>>>(
                x, offsets, pack, out);
        }
    } else {
        // Fallback: zero hi then f32 atomic scatter
        int n4 = (NNODES * FDIM) / 4;
        gcn_zero_kernel<<<(n4 + 255) / 256, 256, 0, stream>>>((float4*)out, n4);
        long long thr = (long long)NEDGES * 32;
        gcn_spmm_kernel<<<(int)((thr + 255) / 256), 256, 0, stream>>>(
            x, edge_w, esrc, edst, out);
    }

    // Fused blend + WMMA GEMM + epilogue (in place over d_out)
    gcn_fused_kernel<<<(NNODES + 63) / 64, 128, 0, stream>>>(
        x, h0, weight, lamda, alpha, lval, out);
}